// FCNetRegressionThreshold_1056561955512
// MI455X (gfx1250) — compile-verified
//
#include <hip/hip_runtime.h>

typedef __attribute__((ext_vector_type(2))) float v2f;
typedef __attribute__((ext_vector_type(8))) float v8f;

// ds_swizzle_b32, group-of-32 mode, XOR pattern: offset = (xor<<10) | 0x1F
template <int IMM>
__device__ __forceinline__ float swz(float v) {
    return __builtin_bit_cast(
        float, __builtin_amdgcn_ds_swizzle(__builtin_bit_cast(int, v), IMM));
}

// Combine row-vector a (lanes where sel==0) and b (sel==1) while reducing
// across xor-distance D: result(lane) = rowX(lane) + rowX(lane^D).
template <int IMM>
__device__ __forceinline__ float merge_rows(float a, float b, bool sel) {
    float t = sel ? b : a;
    float u = sel ? a : b;
    return t + swz<IMM>(u);
}

#define SWZ_X1 0x041F
#define SWZ_X2 0x081F
#define SWZ_X4 0x101F
#define SWZ_X8 0x201F

#define TILES_PER_WAVE 8

// One wave32 processes TILES_PER_WAVE consecutive 16-row tiles:
//   layer1: 16x10 @ 10x16 via 3x V_WMMA_F32_16X16X4_F32 (K padded 10->12),
//           with b1 folded into the WMMA accumulator (D = A*B + C, C = b1)
//   layer2: 16-wide dot per row via a merge-tree of 8 ds_swizzle steps
//   epilogue: relu + threshold-round + reversed store (once per row).
__global__ __launch_bounds__(256) void fcnet_threshold_kernel(
    const float* __restrict__ x,    // [B,10]
    const float* __restrict__ W1,   // [10,16] row-major (k,n)
    const float* __restrict__ b1,   // [16]
    const float* __restrict__ W2,   // [16,1]
    const float* __restrict__ b2,   // [1]
    const float* __restrict__ thr,  // [1]
    float* __restrict__ out,        // [B] (reversed order)
    int B)
{
    const int lane = threadIdx.x & 31;
    const int wave = threadIdx.x >> 5;
    const long wid = (long)blockIdx.x * 8 + wave;   // wave id

    const int half = lane >> 4;        // 0: lanes 0-15, 1: lanes 16-31
    const int ln   = lane & 15;        // 0..15 (M for A-load, N for B/C/D)
    const float fpadmask = (half == 0) ? 1.0f : 0.0f;  // zero for K=10,11 lanes

    // ---- wave-invariant: W1 B-tiles (f32 4x16 layout), biases, scalars ----
    // B-tile layout per K=4 chunk kb: VGPR v holds K = kb + v + 2*half, N = ln
    v2f w0, w1, w2t;
    {
        int k0 = 0 + 2 * half;
        w0.x = W1[(k0    ) * 16 + ln];
        w0.y = W1[(k0 + 1) * 16 + ln];
        int k1 = 4 + 2 * half;
        w1.x = W1[(k1    ) * 16 + ln];
        w1.y = W1[(k1 + 1) * 16 + ln];
        // chunk 2: K = 8,9 valid, K = 10,11 are zero padding (unconditional
        // in-bounds loads, then mask -> no exec-mask branches)
        w2t.x = W1[8 * 16 + ln] * fpadmask;
        w2t.y = W1[9 * 16 + ln] * fpadmask;
    }
    const float b1n   = b1[ln];      // N = ln  -> C-matrix init (bias fold)
    const float w2n   = W2[ln];      // W2[n,0]
    const float bias2 = b2[0];
    const float t     = thr[0];

    const bool sel1 = (lane & 1) != 0;
    const bool sel2 = (lane & 2) != 0;
    const bool sel4 = (lane & 4) != 0;
    const int  m    = (lane & 7) + half * 8;   // output row this lane owns
    const bool writer = (lane & 8) == 0;

    #pragma unroll
    for (int it = 0; it < TILES_PER_WAVE; ++it) {
        const long base = (wid * TILES_PER_WAVE + it) * 16;
        if (base >= (long)B) break;            // wave-uniform

        // ---- A tile: x rows, f32 16x4 layout ----
        // VGPR v holds K = kb + v + 2*half, M = ln
        const float* xr = x + (base + ln) * 10;
        v2f a0 = *(const v2f*)(xr + 0 + 2 * half);   // K 0..3
        v2f a1 = *(const v2f*)(xr + 4 + 2 * half);   // K 4..7
        v2f a2;                                       // K 8..11 (10,11 pad)
        a2.x = xr[8] * fpadmask;
        a2.y = xr[9] * fpadmask;

        // ---- layer 1 GEMM: 3 chained f32 WMMAs, C initialized to b1 ----
        v8f c = {b1n, b1n, b1n, b1n, b1n, b1n, b1n, b1n};
        c = __builtin_amdgcn_wmma_f32_16x16x4_f32(false, a0, false, w0,
                                                  (short)0, c, false, false);
        c = __builtin_amdgcn_wmma_f32_16x16x4_f32(false, a1, false, w1,
                                                  (short)0, c, false, false);
        c = __builtin_amdgcn_wmma_f32_16x16x4_f32(false, a2, false, w2t,
                                                  (short)0, c, false, false);

        // ---- layer 2: s = relu(h) * W2[n]  (h already includes b1) ----
        // C/D layout: VGPR v holds row M = v + 8*half at column N = ln
        float s0 = fmaxf(c[0], 0.0f) * w2n;
        float s1 = fmaxf(c[1], 0.0f) * w2n;
        float s2 = fmaxf(c[2], 0.0f) * w2n;
        float s3 = fmaxf(c[3], 0.0f) * w2n;
        float s4 = fmaxf(c[4], 0.0f) * w2n;
        float s5 = fmaxf(c[5], 0.0f) * w2n;
        float s6 = fmaxf(c[6], 0.0f) * w2n;
        float s7 = fmaxf(c[7], 0.0f) * w2n;

        // ---- merge-tree reduction: 8 rows x 16 lanes -> row sum per lane ----
        float t01 = merge_rows<SWZ_X1>(s0, s1, sel1);
        float t23 = merge_rows<SWZ_X1>(s2, s3, sel1);
        float t45 = merge_rows<SWZ_X1>(s4, s5, sel1);
        float t67 = merge_rows<SWZ_X1>(s6, s7, sel1);
        float t03 = merge_rows<SWZ_X2>(t01, t23, sel2);
        float t47 = merge_rows<SWZ_X2>(t45, t67, sel2);
        float t07 = merge_rows<SWZ_X4>(t03, t47, sel4);
        float tot = t07 + swz<SWZ_X8>(t07);
        // lane l now holds full sum for row m = (l&7) + 8*(l>=16)

        // ---- epilogue: relu + threshold + reversed store ----
        float y = tot + bias2;
        y = fmaxf(y, 0.0f);
        float d = y - truncf(y);
        float r = (d > t) ? rintf(y) : floorf(y);   // jnp.round == rint (RNE)
        if (writer) {
            out[(long)B - 1 - (base + m)] = r;      // reversed output order
        }
    }
}

extern "C" void kernel_launch(void* const* d_in, const int* in_sizes, int n_in,
                              void* d_out, int out_size, void* d_ws, size_t ws_size,
                              hipStream_t stream) {
    const float* x   = (const float*)d_in[0];
    const float* W1  = (const float*)d_in[1];
    const float* b1  = (const float*)d_in[2];
    const float* W2  = (const float*)d_in[3];
    const float* b2  = (const float*)d_in[4];
    const float* thr = (const float*)d_in[5];
    float* out = (float*)d_out;

    const int B = in_sizes[0] / 10;                       // rows of x
    const int tiles = (B + 15) / 16;                      // 16-row tiles
    const int waves = (tiles + TILES_PER_WAVE - 1) / TILES_PER_WAVE;
    const int blocks = (waves + 7) / 8;                   // 8 waves per block
    fcnet_threshold_kernel<<<blocks, 256, 0, stream>>>(x, W1, b1, W2, b2, thr,
                                                       out, B);
}